// LBFGSBlindPnP_19413252178440
// MI455X (gfx1250) — compile-verified
//
#include <hip/hip_runtime.h>
#include <cmath>

typedef float v2f __attribute__((ext_vector_type(2)));
typedef float v8f __attribute__((ext_vector_type(8)));

#define HIST 8
#define DIM 6
#define C1F 0.0001f
#define MAX_LS 25

// ---------------------------------------------------------------------------
// Phase 1: A[b][j][c] = sum_i W[b][i][j] * p2d[b][i][c]   (c = 0..2)
//
// One wave per (batch, 16-wide j tile), WMMA f32 16x16x4, D = A x B + C with
//   A (16x4, MxK)  = p2d^T chunk : A[M][K] = p2d[i+K][M], rows M>=3 zero
//                    lane layout: M = lane&15, K = (lane>>4)*2 + v
//   B (4x16, KxN)  = W chunk     : B[K][N] = W[i+K][j0+N]
//                    lane layout: N = lane&15, K = (lane>>4)*2 + v
//                    -> per-lane loads of W are 16 contiguous dwords per row
//   D (16x16)      : D[M][N] = sum_i p2d[i][M] * W[i][j0+N]
//                    lane layout: M = v + (lane>>4)*8, N = lane&15
//                    -> lanes 0..15, acc[0..2] = A-out[j0+lane][0..2]
// ---------------------------------------------------------------------------
__global__ __launch_bounds__(32)
void wtp_wmma_kernel(const float* __restrict__ W, const float* __restrict__ p2d,
                     float* __restrict__ A, int m, int n)
{
    const int lane  = threadIdx.x & 31;
    const int numJT = (n + 15) >> 4;
    const int bidx  = blockIdx.x / numJT;
    const int j0    = (blockIdx.x % numJT) << 4;

    const float* Wb = W   + (size_t)bidx * m * n;
    const float* Pb = p2d + (size_t)bidx * m * 3;

    const int  jl = lane & 15;
    const int  hi = lane >> 4;
    const int  kb = hi * 2;             // K base for this half-wave

    // A-fragment (p2d): only components 0..2 are real rows
    const bool cok = (jl < 3);
    const int  cc  = cok ? jl : 0;

    // B-fragment (W): column j within the tile
    const int  j   = j0 + jl;
    const bool jok = (j < n);
    const int  jc  = jok ? j : 0;       // clamp: never touch OOB addresses

    v8f acc = {};
    if (j0 + 16 <= n) {
        // Fast path (62 of 63 tiles): W loads fully unmasked.
#pragma unroll 4
        for (int i = 0; i < m; i += 4) {          // m assumed multiple of 4
            const float* pp = Pb + (size_t)(i + kb) * 3 + cc;
            float a0 = pp[0];
            float a1 = pp[3];
            v2f af;
            af.x = cok ? a0 : 0.0f;
            af.y = cok ? a1 : 0.0f;

            const float* wp = Wb + (size_t)(i + kb) * n + j;
            v2f bf;
            bf.x = wp[0];
            bf.y = wp[n];

            acc = __builtin_amdgcn_wmma_f32_16x16x4_f32(false, af, false, bf,
                                                        (short)0, acc, false, false);
        }
    } else {
        // Tail tile: mask W columns j >= n to zero.
#pragma unroll 2
        for (int i = 0; i < m; i += 4) {
            const float* pp = Pb + (size_t)(i + kb) * 3 + cc;
            float a0 = pp[0];
            float a1 = pp[3];
            v2f af;
            af.x = cok ? a0 : 0.0f;
            af.y = cok ? a1 : 0.0f;

            const float* wp = Wb + (size_t)(i + kb) * n + jc;
            float w0 = wp[0];
            float w1 = wp[n];
            v2f bf;
            bf.x = jok ? w0 : 0.0f;
            bf.y = jok ? w1 : 0.0f;

            acc = __builtin_amdgcn_wmma_f32_16x16x4_f32(false, af, false, bf,
                                                        (short)0, acc, false, false);
        }
    }

    // Store: lanes 0..15 each own one output point j, components in acc[0..2].
    if (hi == 0 && jok) {
        float* Ao = A + ((size_t)bidx * n + j) * 3;
        Ao[0] = acc[0];
        Ao[1] = acc[1];
        Ao[2] = acc[2];
    }
}

// ---------------------------------------------------------------------------
// Phase 2: L-BFGS, one block (256 threads) per batch element.
// f(theta) = const - sum_j a_j . normalize(R p3d_j + t); the constant cancels
// in both the gradient and the Armijo test, so it is dropped.
// ---------------------------------------------------------------------------
__device__ __forceinline__ float block_reduce_sum(float v)
{
    __shared__ float red[256];
    const int t = threadIdx.x;
    __syncthreads();
    red[t] = v;
    __syncthreads();
    for (int s = 128; s > 0; s >>= 1) {
        if (t < s) red[t] += red[t + s];
        __syncthreads();
    }
    return red[0];
}

__device__ __forceinline__ void aa_to_R(const float* t, float* R)
{
    float rx = t[0], ry = t[1], rz = t[2];
    float th2 = rx * rx + ry * ry + rz * rz;
    float th  = sqrtf(th2 + 1e-12f);
    float inv = 1.0f / th;
    float kx = rx * inv, ky = ry * inv, kz = rz * inv;
    float s = sinf(th), c = cosf(th), oc = 1.0f - c;
    R[0] = 1.0f - oc * (ky * ky + kz * kz);
    R[1] = -s * kz + oc * (kx * ky);
    R[2] =  s * ky + oc * (kx * kz);
    R[3] =  s * kz + oc * (kx * ky);
    R[4] = 1.0f - oc * (kx * kx + kz * kz);
    R[5] = -s * kx + oc * (ky * kz);
    R[6] = -s * ky + oc * (kx * kz);
    R[7] =  s * kx + oc * (ky * kz);
    R[8] = 1.0f - oc * (kx * kx + ky * ky);
}

__device__ float eval_f(const float* th, const float* Ac, const float* P, int n)
{
    float R[9];
    aa_to_R(th, R);
    float acc = 0.0f;
    for (int j = threadIdx.x; j < n; j += blockDim.x) {
        float px = P[j * 3 + 0], py = P[j * 3 + 1], pz = P[j * 3 + 2];
        float ux = R[0] * px + R[1] * py + R[2] * pz + th[3];
        float uy = R[3] * px + R[4] * py + R[5] * pz + th[4];
        float uz = R[6] * px + R[7] * py + R[8] * pz + th[5];
        float nu = sqrtf(ux * ux + uy * uy + uz * uz) + 1e-12f;
        float ax = Ac[j * 3 + 0], ay = Ac[j * 3 + 1], az = Ac[j * 3 + 2];
        acc += (ax * ux + ay * uy + az * uz) / nu;
    }
    return -block_reduce_sum(acc);
}

// Analytic gradient (Gallego-Yezzi): d(Rp)/dr_i = (r_i (r x v) + (r x (e_i - c_i)) x v)/|r|^2
__device__ void eval_grad(const float* th, const float* Ac, const float* P, int n, float* g)
{
    float R[9];
    aa_to_R(th, R);
    float rx = th[0], ry = th[1], rz = th[2];
    float inv_th2 = 1.0f / (rx * rx + ry * ry + rz * rz + 1e-12f);
    float w[3][3];
#pragma unroll
    for (int i = 0; i < 3; ++i) {
        float ex = (i == 0 ? 1.0f : 0.0f) - R[0 + i];
        float ey = (i == 1 ? 1.0f : 0.0f) - R[3 + i];
        float ez = (i == 2 ? 1.0f : 0.0f) - R[6 + i];
        w[i][0] = ry * ez - rz * ey;
        w[i][1] = rz * ex - rx * ez;
        w[i][2] = rx * ey - ry * ex;
    }
    float gr0 = 0.f, gr1 = 0.f, gr2 = 0.f, gt0 = 0.f, gt1 = 0.f, gt2 = 0.f;
    for (int j = threadIdx.x; j < n; j += blockDim.x) {
        float px = P[j * 3 + 0], py = P[j * 3 + 1], pz = P[j * 3 + 2];
        float vx = R[0] * px + R[1] * py + R[2] * pz;
        float vy = R[3] * px + R[4] * py + R[5] * pz;
        float vz = R[6] * px + R[7] * py + R[8] * pz;
        float ux = vx + th[3], uy = vy + th[4], uz = vz + th[5];
        float nu = sqrtf(ux * ux + uy * uy + uz * uz);
        float invd = 1.0f / (nu + 1e-12f);
        float ax = Ac[j * 3 + 0], ay = Ac[j * 3 + 1], az = Ac[j * 3 + 2];
        float adotu = ax * ux + ay * uy + az * uz;
        float coef = adotu * invd * invd / fmaxf(nu, 1e-20f);
        float gux = -ax * invd + coef * ux;
        float guy = -ay * invd + coef * uy;
        float guz = -az * invd + coef * uz;
        gt0 += gux; gt1 += guy; gt2 += guz;
        float cx = ry * vz - rz * vy;
        float cy = rz * vx - rx * vz;
        float cz = rx * vy - ry * vx;
#pragma unroll
        for (int i = 0; i < 3; ++i) {
            float wxx = w[i][1] * vz - w[i][2] * vy;
            float wxy = w[i][2] * vx - w[i][0] * vz;
            float wxz = w[i][0] * vy - w[i][1] * vx;
            float ri = (i == 0) ? rx : ((i == 1) ? ry : rz);
            float dx = (ri * cx + wxx) * inv_th2;
            float dy = (ri * cy + wxy) * inv_th2;
            float dz = (ri * cz + wxz) * inv_th2;
            float contrib = gux * dx + guy * dy + guz * dz;
            if (i == 0) gr0 += contrib; else if (i == 1) gr1 += contrib; else gr2 += contrib;
        }
    }
    g[0] = block_reduce_sum(gr0);
    g[1] = block_reduce_sum(gr1);
    g[2] = block_reduce_sum(gr2);
    g[3] = block_reduce_sum(gt0);
    g[4] = block_reduce_sum(gt1);
    g[5] = block_reduce_sum(gt2);
}

__global__ __launch_bounds__(256)
void lbfgs_kernel(const float* __restrict__ Aall, const float* __restrict__ p3d,
                  const float* __restrict__ pose0, float* __restrict__ out,
                  int n, int num_iter)
{
    const int b = blockIdx.x;
    const float* Ac = Aall + (size_t)b * n * 3;
    const float* P  = p3d  + (size_t)b * n * 3;

    __shared__ float Ssm[HIST][DIM], Ysm[HIST][DIM], rhosm[HIST];
    for (int i = threadIdx.x; i < HIST * DIM; i += blockDim.x) {
        (&Ssm[0][0])[i] = 0.0f;
        (&Ysm[0][0])[i] = 0.0f;
    }
    if (threadIdx.x < HIST) rhosm[threadIdx.x] = 0.0f;
    __syncthreads();

    float x[DIM];
#pragma unroll
    for (int i = 0; i < DIM; ++i) x[i] = pose0[b * DIM + i];

    float fx = eval_f(x, Ac, P, n);
    float g[DIM];
    eval_grad(x, Ac, P, n, g);

    for (int k = 0; k < num_iter; ++k) {
        // two-loop recursion (uniform across threads; reductions are identical)
        float q[DIM];
#pragma unroll
        for (int i = 0; i < DIM; ++i) q[i] = g[i];
        float al[HIST];
        int idxs[HIST];
#pragma unroll
        for (int i = 0; i < HIST; ++i) {
            int idx = ((k - 1 - i) % HIST + HIST) % HIST;
            float a = 0.0f;
#pragma unroll
            for (int jj = 0; jj < DIM; ++jj) a += Ssm[idx][jj] * q[jj];
            a *= rhosm[idx];
#pragma unroll
            for (int jj = 0; jj < DIM; ++jj) q[jj] -= a * Ysm[idx][jj];
            al[i] = a;
            idxs[i] = idx;
        }
        int idx0 = ((k - 1) % HIST + HIST) % HIST;
        float sy0 = 0.0f, yy0 = 0.0f;
#pragma unroll
        for (int jj = 0; jj < DIM; ++jj) {
            sy0 += Ssm[idx0][jj] * Ysm[idx0][jj];
            yy0 += Ysm[idx0][jj] * Ysm[idx0][jj];
        }
        float gamma = (k > 0) ? sy0 / (yy0 + 1e-10f) : 1.0f;
        float d[DIM];
#pragma unroll
        for (int jj = 0; jj < DIM; ++jj) d[jj] = gamma * q[jj];
#pragma unroll
        for (int i = HIST - 1; i >= 0; --i) {
            int idx = idxs[i];
            float bfac = 0.0f;
#pragma unroll
            for (int jj = 0; jj < DIM; ++jj) bfac += Ysm[idx][jj] * d[jj];
            bfac *= rhosm[idx];
#pragma unroll
            for (int jj = 0; jj < DIM; ++jj) d[jj] += Ssm[idx][jj] * (al[i] - bfac);
        }
        float gd = 0.0f;
#pragma unroll
        for (int jj = 0; jj < DIM; ++jj) { d[jj] = -d[jj]; gd += g[jj] * d[jj]; }
        if (!(gd < 0.0f)) {
            gd = 0.0f;
#pragma unroll
            for (int jj = 0; jj < DIM; ++jj) { d[jj] = -g[jj]; gd -= g[jj] * g[jj]; }
        }

        // Armijo backtracking (uniform loop; matches reference while_loop)
        float t = 1.0f;
        float xt[DIM];
#pragma unroll
        for (int jj = 0; jj < DIM; ++jj) xt[jj] = x[jj] + d[jj];
        float fn = eval_f(xt, Ac, P, n);
        int c = 0;
        while (!(fn <= fx + C1F * t * gd) && c < MAX_LS) {
            t *= 0.5f;
#pragma unroll
            for (int jj = 0; jj < DIM; ++jj) xt[jj] = x[jj] + t * d[jj];
            fn = eval_f(xt, Ac, P, n);
            ++c;
        }

        float gn[DIM];
        eval_grad(xt, Ac, P, n, gn);
        float sv[DIM], yv[DIM], sy = 0.0f;
#pragma unroll
        for (int jj = 0; jj < DIM; ++jj) {
            sv[jj] = xt[jj] - x[jj];
            yv[jj] = gn[jj] - g[jj];
            sy += sv[jj] * yv[jj];
        }
        float rn = (sy > 1e-10f) ? 1.0f / sy : 0.0f;
        int slot = k % HIST;
        __syncthreads();
        if (threadIdx.x == 0) {
#pragma unroll
            for (int jj = 0; jj < DIM; ++jj) { Ssm[slot][jj] = sv[jj]; Ysm[slot][jj] = yv[jj]; }
            rhosm[slot] = rn;
        }
        __syncthreads();
#pragma unroll
        for (int jj = 0; jj < DIM; ++jj) { x[jj] = xt[jj]; g[jj] = gn[jj]; }
        fx = fn;
    }

    if (threadIdx.x == 0) {
#pragma unroll
        for (int jj = 0; jj < DIM; ++jj) out[b * DIM + jj] = x[jj];
    }
}

extern "C" void kernel_launch(void* const* d_in, const int* in_sizes, int n_in,
                              void* d_out, int out_size, void* d_ws, size_t ws_size,
                              hipStream_t stream)
{
    const float* W    = (const float*)d_in[0];
    const float* p2d  = (const float*)d_in[1];
    const float* p3d  = (const float*)d_in[2];
    const float* pose = (const float*)d_in[3];

    const int b = in_sizes[3] / 6;
    const int m = in_sizes[1] / (b * 3);
    const int n = in_sizes[2] / (b * 3);

    float* A = (float*)d_ws;                       // (b, n, 3) f32 scratch

    const int numJT = (n + 15) / 16;
    wtp_wmma_kernel<<<b * numJT, 32, 0, stream>>>(W, p2d, A, m, n);

    const int max_pts = (m > n) ? m : n;
    double it_d = 75.0 * pow((double)max_pts / 1000.0, -1.5);
    if (it_d > 100.0) it_d = 100.0;
    if (it_d < 1.0) it_d = 1.0;
    const int num_iter = (int)llround(it_d);

    lbfgs_kernel<<<b, 256, 0, stream>>>(A, p3d, pose, (float*)d_out, n, num_iter);
}